// DomainSpecificAttention_77120432767310
// MI455X (gfx1250) — compile-verified
//
#include <hip/hip_runtime.h>
#include <hip/hip_bf16.h>
#include <math.h>

typedef float v2f __attribute__((ext_vector_type(2)));
typedef float v8f __attribute__((ext_vector_type(8)));

#define PI_F 3.14159265358979f

// ---------------- model dims ----------------
constexpr int BB = 2, HH = 48, WW = 48, CC = 3, DD = 64, NHEAD = 8;
constexpr int TT = HH * WW;            // 2304 tokens
constexpr int BHH = BB * NHEAD;        // 16 (b,head) pairs
constexpr int QTILES = TT / 16;        // 144

// ---------------- workspace layout (floats) ----------------
constexpr size_t N_XC   = (size_t)BB * TT * DD;        // 294912
constexpr size_t N_QKV  = (size_t)BHH * TT * DD;       // 2359296
constexpr size_t N_IMG3 = (size_t)BB * HH * WW * CC;   // 13824
constexpr size_t N_IMG64= (size_t)BB * HH * WW * 64;   // 294912
constexpr size_t SZ1 = (size_t)BB * 24 * 24 * 3;       // 3456
constexpr size_t SZ2 = (size_t)BB * 12 * 12 * 3;       // 864
constexpr size_t SZ3 = (size_t)BB * 6 * 6 * 3;         // 216

constexpr size_t O_XC  = 0;
constexpr size_t O_Q   = O_XC + N_XC;
constexpr size_t O_K   = O_Q + N_QKV;
constexpr size_t O_V   = O_K + N_QKV;
constexpr size_t O_AO  = O_V + N_QKV;
constexpr size_t O_DCT1= O_AO + N_QKV;
constexpr size_t O_DCT2= O_DCT1 + N_IMG3;
constexpr size_t O_BW  = O_DCT2 + N_IMG3;   // 16
constexpr size_t O_PW  = O_BW + 16;         // 2 (+pad)
constexpr size_t O_FCN = O_PW + 4;
constexpr size_t O_FDT = O_FCN + N_IMG64;
constexpr size_t O_A1  = O_FDT + N_IMG64;
constexpr size_t O_H1  = O_A1 + SZ1;
constexpr size_t O_V1  = O_H1 + SZ1;
constexpr size_t O_D1  = O_V1 + SZ1;
constexpr size_t O_A2  = O_D1 + SZ1;
constexpr size_t O_H2  = O_A2 + SZ2;
constexpr size_t O_V2  = O_H2 + SZ2;
constexpr size_t O_D2  = O_V2 + SZ2;
constexpr size_t O_A3  = O_D2 + SZ2;
constexpr size_t O_H3  = O_A3 + SZ3;
constexpr size_t O_V3  = O_H3 + SZ3;
constexpr size_t O_D3  = O_V3 + SZ3;
constexpr size_t O_DW3 = O_D3 + SZ3;        // 18
constexpr size_t O_DW2 = O_DW3 + 18;
constexpr size_t O_DW1 = O_DW2 + 18;
constexpr size_t O_R6  = O_DW1 + 18;
constexpr size_t O_R12 = O_R6 + SZ3;
constexpr size_t O_R24 = O_R12 + SZ2;
// total ws: ~10.4M floats (~42 MB) — assumed <= ws_size

// attention LDS: Q tile (16x64) + score row-block (16x2304) + reductions
constexpr int ATTN_LDS_FLOATS = 16 * 64 + 16 * TT + 128 + 16;
constexpr int ATTN_SMEM_BYTES = ATTN_LDS_FLOATS * 4;   // 152128 B (< 320KB/WGP)

__device__ __forceinline__ float dct2v(int k, int n) {
  return 2.f * __cosf(PI_F * (2.f * n + 1.f) * (float)k / 96.f);
}
__device__ __forceinline__ float dct3v(int k, int n) {
  return (n == 0) ? 1.f : 2.f * __cosf(PI_F * (float)n * (2.f * k + 1.f) / 96.f);
}
__device__ __forceinline__ float sigm(float x) { return 1.f / (1.f + __expf(-x)); }

// ---------------- spatial: xc, pos, q/k/v projections ----------------
__global__ __launch_bounds__(64) void k_qkv(
    const float* __restrict__ x, const float* __restrict__ in_w, const float* __restrict__ in_b,
    const float* __restrict__ pos_w, const float* __restrict__ pos_b,
    const float* __restrict__ wq, const float* __restrict__ bq,
    const float* __restrict__ wk, const float* __restrict__ bk,
    const float* __restrict__ wv, const float* __restrict__ bv,
    float* __restrict__ xc, float* __restrict__ q, float* __restrict__ k, float* __restrict__ v)
{
  __shared__ float xcs[64], poss[64];
  int b = blockIdx.x / TT, t = blockIdx.x % TT;
  int d = threadIdx.x;
  const float* xp = x + ((size_t)b * TT + t) * 3;
  float acc = in_b[d];
  for (int c = 0; c < 3; ++c) acc += xp[c] * in_w[c * 64 + d];
  xcs[d] = acc;
  xc[((size_t)b * TT + t) * 64 + d] = acc;
  __syncthreads();
  float p = pos_b[d];
  for (int e = 0; e < 64; ++e) p += xcs[e] * pos_w[e * 64 + d];
  poss[d] = p;
  __syncthreads();
  for (int j = 0; j < 8; ++j) {
    int he = d + j * 64;
    int h = he >> 6, e = he & 63;
    float aq = bq[he], ak = bk[he], av = bv[he];
    for (int dd = 0; dd < 64; ++dd) {
      aq += poss[dd] * wq[dd * 512 + he];
      ak += xcs[dd]  * wk[dd * 512 + he];
      av += xcs[dd]  * wv[dd * 512 + he];
    }
    size_t oidx = ((size_t)(b * 8 + h) * TT + t) * 64 + e;
    q[oidx] = aq; k[oidx] = ak; v[oidx] = av;
  }
}

// ---------------- flash-style attention, fp32 WMMA 16x16x4 ----------------
// one workgroup per (b,head,16-row q block); full score row-block lives in LDS.
__global__ __launch_bounds__(128) void k_attn(const float* __restrict__ q,
                                              const float* __restrict__ kg,
                                              const float* __restrict__ vg,
                                              float* __restrict__ o)
{
  extern __shared__ float smem[];
  float* Qs = smem;                  // [16][64], pre-scaled by 1/sqrt(E)
  float* Ss = smem + 16 * 64;        // [16][T]
  float* red = Ss + 16 * TT;         // [16][8]
  float* rowstat = red + 128;        // [16]

  int bh = blockIdx.x / QTILES;
  int q0 = (blockIdx.x % QTILES) * 16;
  int tid = threadIdx.x;
  int lane = tid & 31, wave = tid >> 5;
  int hl = lane >> 4, l16 = lane & 15;   // half-wave split per WMMA layout

  const float* Qg = q  + (size_t)bh * TT * 64;
  const float* Kg = kg + (size_t)bh * TT * 64;
  const float* Vg = vg + (size_t)bh * TT * 64;

  for (int i = tid; i < 16 * 64; i += 128)
    Qs[i] = Qg[(size_t)(q0 + (i >> 6)) * 64 + (i & 63)] * 0.125f;
  __syncthreads();

  // phase 1: S = (Q/8) K^T, 16x16 tiles, waves stride over s-tiles
  for (int st = wave; st < QTILES; st += 4) {
    int s0 = st * 16;
    v8f acc = {};
#pragma unroll
    for (int k0 = 0; k0 < 64; k0 += 4) {
      int ka = k0 + 2 * hl;
      v2f a, bm;
      a.x = Qs[l16 * 64 + ka];  a.y = Qs[l16 * 64 + ka + 1];
      const float* kr = Kg + (size_t)(s0 + l16) * 64 + ka;   // B[k][n] = K[n][k]
      bm.x = kr[0]; bm.y = kr[1];
      acc = __builtin_amdgcn_wmma_f32_16x16x4_f32(false, a, false, bm, (short)0, acc,
                                                  false, false);
    }
#pragma unroll
    for (int i = 0; i < 8; ++i)
      Ss[(i + hl * 8) * TT + s0 + l16] = acc[i];
  }
  __syncthreads();

  // softmax over S rows (8 threads per row)
  int r = tid >> 3, sub = tid & 7;
  float mx = -3.0e38f;
  for (int c = sub; c < TT; c += 8) mx = fmaxf(mx, Ss[r * TT + c]);
  red[r * 8 + sub] = mx;
  __syncthreads();
  if (sub == 0) {
    float m2 = red[r * 8];
    for (int j = 1; j < 8; ++j) m2 = fmaxf(m2, red[r * 8 + j]);
    rowstat[r] = m2;
  }
  __syncthreads();
  float rm = rowstat[r], sm = 0.f;
  for (int c = sub; c < TT; c += 8) {
    float e = __expf(Ss[r * TT + c] - rm);
    Ss[r * TT + c] = e; sm += e;
  }
  red[r * 8 + sub] = sm;
  __syncthreads();
  if (sub == 0) {
    float s2 = 0.f;
    for (int j = 0; j < 8; ++j) s2 += red[r * 8 + j];
    rowstat[r] = s2;
  }
  __syncthreads();
  float inv = 1.f / rowstat[r];
  for (int c = sub; c < TT; c += 8) Ss[r * TT + c] *= inv;
  __syncthreads();

  // phase 2: O = P V, each wave owns one 16-wide slice of E
  int n0 = wave * 16;
  v8f acc = {};
  for (int k0 = 0; k0 < TT; k0 += 4) {
    int ka = k0 + 2 * hl;
    v2f a, bm;
    a.x = Ss[l16 * TT + ka];  a.y = Ss[l16 * TT + ka + 1];
    const float* vr = Vg + (size_t)ka * 64 + n0 + l16;       // B[k][n] = V[k][n]
    bm.x = vr[0]; bm.y = vr[64];
    acc = __builtin_amdgcn_wmma_f32_16x16x4_f32(false, a, false, bm, (short)0, acc,
                                                false, false);
  }
  float* Ob = o + (size_t)bh * TT * 64;
#pragma unroll
  for (int i = 0; i < 8; ++i)
    Ob[(size_t)(q0 + i + hl * 8) * 64 + n0 + l16] = acc[i];
}

// ---------------- output proj + residual + double LN -> spatial ----------------
__global__ __launch_bounds__(64) void k_oproj(
    const float* __restrict__ o, const float* __restrict__ wo, const float* __restrict__ bo,
    const float* __restrict__ xc,
    const float* __restrict__ g1, const float* __restrict__ b1,
    const float* __restrict__ g2, const float* __restrict__ b2,
    float* __restrict__ spat)
{
  __shared__ float s1[64], s2[64];
  int b = blockIdx.x / TT, t = blockIdx.x % TT;
  int d = threadIdx.x;
  float acc = bo[d];
  for (int h = 0; h < 8; ++h) {
    const float* ob = o + ((size_t)(b * 8 + h) * TT + t) * 64;
    const float* wb = wo + h * 4096 + d;
    for (int e = 0; e < 64; ++e) acc += ob[e] * wb[e * 64];
  }
  float val = xc[((size_t)b * TT + t) * 64 + d] + acc;
  s1[d] = val; s2[d] = val * val; __syncthreads();
  for (int st = 32; st > 0; st >>= 1) {
    if (d < st) { s1[d] += s1[d + st]; s2[d] += s2[d + st]; }
    __syncthreads();
  }
  float m = s1[0] * (1.f / 64.f);
  float var = s2[0] * (1.f / 64.f) - m * m;
  float v1 = (val - m) * rsqrtf(var + 1e-3f) * g1[d] + b1[d];
  __syncthreads();
  s1[d] = v1; s2[d] = v1 * v1; __syncthreads();
  for (int st = 32; st > 0; st >>= 1) {
    if (d < st) { s1[d] += s1[d + st]; s2[d] += s2[d + st]; }
    __syncthreads();
  }
  m = s1[0] * (1.f / 64.f);
  var = s2[0] * (1.f / 64.f) - m * m;
  spat[((size_t)b * TT + t) * 64 + d] = (v1 - m) * rsqrtf(var + 1e-3f) * g2[d] + b2[d];
}

// ---------------- frequency branch ----------------
__global__ void k_dct2_w(const float* __restrict__ in, float* __restrict__ out)
{
  int idx = blockIdx.x * 256 + threadIdx.x;
  if (idx >= BB * 48 * 48 * 3) return;
  int c = idx % 3; int tmp = idx / 3;
  int kk = tmp % 48; tmp /= 48;
  int h = tmp % 48; int b = tmp / 48;
  const float* xr = in + (((size_t)b * 48 + h) * 48) * 3 + c;
  float acc = 0.f;
  for (int w = 0; w < 48; ++w) acc += xr[w * 3] * dct2v(kk, w);
  out[idx] = acc;
}

__global__ void k_dct2_h(const float* __restrict__ in, float* __restrict__ out)
{
  int idx = blockIdx.x * 256 + threadIdx.x;
  if (idx >= BB * 48 * 48 * 3) return;
  int c = idx % 3; int tmp = idx / 3;
  int w = tmp % 48; tmp /= 48;
  int kk = tmp % 48; int b = tmp / 48;
  const float* ir = in + ((size_t)b * 48 * 48) * 3 + w * 3 + c;
  float acc = 0.f;
  for (int h = 0; h < 48; ++h) acc += ir[(size_t)h * 48 * 3] * dct2v(kk, h);
  out[idx] = acc;
}

__global__ __launch_bounds__(256) void k_band(
    const float* __restrict__ x, const float* __restrict__ d2,
    const float* __restrict__ band_w, const float* __restrict__ band_b,
    const float* __restrict__ phase_w, const float* __restrict__ phase_b,
    float* __restrict__ bw_out, float* __restrict__ pw_out)
{
  __shared__ float fr[24 * 8];
  __shared__ float pr[256];
  __shared__ float feats[24];
  int b = blockIdx.x;
  int tid = threadIdx.x;
  int ou = tid >> 3, sub = tid & 7;
  if (ou < 24) {
    int band = ou / 3, c = ou % 3;
    float acc = 0.f;
    for (int p = sub; p < 288; p += 8) {
      int r = p / 48, w = p % 48;
      acc += d2[(((size_t)b * 48 + band * 6 + r) * 48 + w) * 3 + c];
    }
    fr[ou * 8 + sub] = acc;
  }
  float pacc = 0.f;
  for (int i = tid; i < 6912; i += 256) {
    float xv = x[(size_t)b * 6912 + i];
    if (xv < 0.f) pacc += PI_F * phase_w[i];
  }
  pr[tid] = pacc;
  __syncthreads();
  for (int st = 128; st > 0; st >>= 1) { if (tid < st) pr[tid] += pr[tid + st]; __syncthreads(); }
  if (tid < 24) {
    float s = 0.f;
    for (int j = 0; j < 8; ++j) s += fr[tid * 8 + j];
    feats[tid] = s * (1.f / 288.f);
  }
  __syncthreads();
  if (tid == 0) {
    pw_out[b] = sigm(pr[0] + phase_b[0]);
    float lg[8], mx = -3e38f;
    for (int j = 0; j < 8; ++j) {
      float l = band_b[j];
      for (int i2 = 0; i2 < 24; ++i2) l += feats[i2] * band_w[i2 * 8 + j];
      lg[j] = l; mx = fmaxf(mx, l);
    }
    float s = 0.f;
    for (int j = 0; j < 8; ++j) { lg[j] = __expf(lg[j] - mx); s += lg[j]; }
    for (int j = 0; j < 8; ++j) bw_out[b * 8 + j] = lg[j] / s;
  }
}

__global__ __launch_bounds__(64) void k_fconv(
    const float* __restrict__ d2, const float* __restrict__ wband, const float* __restrict__ pw,
    const float* __restrict__ cw, const float* __restrict__ cb, float* __restrict__ fout)
{
  int b = blockIdx.x / (48 * 48);
  int ij = blockIdx.x % (48 * 48);
  int i = ij / 48, j = ij % 48;
  int oc = threadIdx.x;
  float acc = cb[oc];
  for (int di = 0; di < 3; ++di) {
    int ii = i + di - 1;
    if (ii < 0 || ii >= 48) continue;
    float wb = wband[b * 8 + ii / 6];
    for (int dj = 0; dj < 3; ++dj) {
      int jj = j + dj - 1;
      if (jj < 0 || jj >= 48) continue;
      const float* dp = d2 + (((size_t)b * 48 + ii) * 48 + jj) * 3;
      const float* wp = cw + ((di * 3 + dj) * 3) * 64 + oc;
      acc += dp[0] * wb * wp[0] + dp[1] * wb * wp[64] + dp[2] * wb * wp[128];
    }
  }
  fout[(((size_t)b * 48 + i) * 48 + j) * 64 + oc] = acc * pw[b];
}

__global__ void k_dct3(const float* __restrict__ in, float* __restrict__ out)
{
  int idx = blockIdx.x * 256 + threadIdx.x;
  if (idx >= BB * 48 * 48 * 64) return;
  int c = idx & 63; int tmp = idx >> 6;
  int kk = tmp % 48; tmp /= 48;
  int h = tmp % 48; int b = tmp / 48;
  const float* ir = in + (((size_t)b * 48 + h) * 48) * 64 + c;
  float acc = 0.f;
  for (int w = 0; w < 48; ++w) acc += ir[(size_t)w * 64] * dct3v(kk, w);
  out[idx] = acc;
}

// ---------------- wavelet branch ----------------
__global__ void k_dwt(const float* __restrict__ in, int h, int w,
                      float* __restrict__ cA, float* __restrict__ cH,
                      float* __restrict__ cV, float* __restrict__ cD)
{
  int h2 = h / 2, w2 = w / 2;
  int N = BB * h2 * w2 * 3;
  int idx = blockIdx.x * 256 + threadIdx.x;
  if (idx >= N) return;
  int c = idx % 3; int tmp = idx / 3;
  int j = tmp % w2; tmp /= w2;
  int i = tmp % h2; int b = tmp / h2;
  size_t r0 = (((size_t)b * h + 2 * i) * w + 2 * j) * 3 + c;
  size_t r1 = r0 + (size_t)w * 3;
  float x00 = in[r0], x01 = in[r0 + 3], x10 = in[r1], x11 = in[r1 + 3];
  cA[idx] = (x00 + x01 + x10 + x11) * 0.5f;
  cH[idx] = (x00 + x01 - x10 - x11) * 0.5f;
  cV[idx] = (x00 - x01 + x10 - x11) * 0.5f;
  cD[idx] = (x00 - x01 - x10 + x11) * 0.5f;
}

__global__ __launch_bounds__(128) void k_appw(
    const float* __restrict__ a3, const float* __restrict__ app_w,
    const float* __restrict__ app_b, float* __restrict__ rec6)
{
  __shared__ float aw[3];
  int b = blockIdx.x, tid = threadIdx.x;
  if (tid < 3) {
    float acc = 0.f;
    for (int p = 0; p < 36; ++p) acc += a3[((size_t)b * 36 + p) * 3 + tid] * app_w[p];
    aw[tid] = sigm(acc + app_b[0]);
  }
  __syncthreads();
  for (int i = tid; i < 108; i += 128)
    rec6[(size_t)b * 108 + i] = a3[(size_t)b * 108 + i] * aw[i % 3];
}

__global__ __launch_bounds__(128) void k_detw(
    const float* __restrict__ cH, const float* __restrict__ cV, const float* __restrict__ cD,
    const float* __restrict__ Wd, const float* __restrict__ bd, int hh, float* __restrict__ dw)
{
  __shared__ float red[3 * 128];
  int b = blockIdx.x / 3, c = blockIdx.x % 3;
  int tid = threadIdx.x;
  int n = hh * hh;
  float a0 = 0.f, a1 = 0.f, a2 = 0.f;
  for (int k2 = 0; k2 < 3; ++k2) {
    const float* det = (k2 == 0) ? cH : ((k2 == 1) ? cV : cD);
    const float* wk2 = Wd + (size_t)k2 * n * 3;
    for (int p = tid; p < n; p += 128) {
      float vv = det[((size_t)b * n + p) * 3 + c];
      a0 += vv * wk2[p * 3 + 0];
      a1 += vv * wk2[p * 3 + 1];
      a2 += vv * wk2[p * 3 + 2];
    }
  }
  red[tid] = a0; red[128 + tid] = a1; red[256 + tid] = a2;
  __syncthreads();
  for (int st = 64; st > 0; st >>= 1) {
    if (tid < st) {
      red[tid] += red[tid + st];
      red[128 + tid] += red[128 + tid + st];
      red[256 + tid] += red[256 + tid + st];
    }
    __syncthreads();
  }
  if (tid == 0) {
    float l0 = red[0] + bd[0], l1 = red[128] + bd[1], l2 = red[256] + bd[2];
    float mx = fmaxf(l0, fmaxf(l1, l2));
    float e0 = __expf(l0 - mx), e1 = __expf(l1 - mx), e2 = __expf(l2 - mx);
    float inv = 1.f / (e0 + e1 + e2);
    dw[(b * 3 + c) * 3 + 0] = e0 * inv;
    dw[(b * 3 + c) * 3 + 1] = e1 * inv;
    dw[(b * 3 + c) * 3 + 2] = e2 * inv;
  }
}

__global__ void k_idwt(const float* __restrict__ rec, const float* __restrict__ cH,
                       const float* __restrict__ cV, const float* __restrict__ cD,
                       const float* __restrict__ dw, int hh, float* __restrict__ out)
{
  int N = BB * hh * hh * 3;
  int idx = blockIdx.x * 256 + threadIdx.x;
  if (idx >= N) return;
  int c = idx % 3; int tmp = idx / 3;
  int j = tmp % hh; tmp /= hh;
  int i = tmp % hh; int b = tmp / hh;
  const float* dwp = dw + (b * 3 + c) * 3;
  float cAv = rec[idx];
  float chv = cH[idx] * dwp[0], cvv = cV[idx] * dwp[1], cdv = cD[idx] * dwp[2];
  float x00 = (cAv + chv + cvv + cdv) * 0.5f;
  float x01 = (cAv + chv - cvv - cdv) * 0.5f;
  float x10 = (cAv - chv + cvv - cdv) * 0.5f;
  float x11 = (cAv - chv - cvv + cdv) * 0.5f;
  int w2 = hh * 2;
  size_t o0 = (((size_t)b * w2 + 2 * i) * w2 + 2 * j) * 3 + c;
  size_t o1 = o0 + (size_t)w2 * 3;
  out[o0] = x00; out[o0 + 3] = x01; out[o1] = x10; out[o1 + 3] = x11;
}

// ---------------- combine: concat conv3x3 -> LN -> conv1x1 ----------------
__global__ __launch_bounds__(64) void k_combine(
    const float* __restrict__ spat, const float* __restrict__ freq, const float* __restrict__ wav,
    const float* __restrict__ fc_w, const float* __restrict__ fc_b,
    const float* __restrict__ fn_g, const float* __restrict__ fn_b,
    const float* __restrict__ oc_w, const float* __restrict__ oc_b,
    float* __restrict__ outp)
{
  __shared__ float s1[64], s2[64], ysh[64];
  int b = blockIdx.x / (48 * 48);
  int ij = blockIdx.x % (48 * 48);
  int i = ij / 48, j = ij % 48;
  int oc = threadIdx.x;
  float acc = fc_b[oc];
  for (int di = 0; di < 3; ++di) {
    int ii = i + di - 1; if (ii < 0 || ii >= 48) continue;
    for (int dj = 0; dj < 3; ++dj) {
      int jj = j + dj - 1; if (jj < 0 || jj >= 48) continue;
      size_t px = ((size_t)b * 48 + ii) * 48 + jj;
      const float* wrow = fc_w + (size_t)((di * 3 + dj) * 131) * 64 + oc;
      const float* sp = spat + px * 64;
      const float* fp = freq + px * 64;
      const float* wp = wav + px * 3;
      for (int ch = 0; ch < 64; ++ch) acc += sp[ch] * wrow[ch * 64];
      for (int ch = 0; ch < 64; ++ch) acc += fp[ch] * wrow[(64 + ch) * 64];
      for (int ch = 0; ch < 3; ++ch)  acc += wp[ch] * wrow[(128 + ch) * 64];
    }
  }
  s1[oc] = acc; s2[oc] = acc * acc; __syncthreads();
  for (int st = 32; st > 0; st >>= 1) {
    if (oc < st) { s1[oc] += s1[oc + st]; s2[oc] += s2[oc + st]; }
    __syncthreads();
  }
  float m = s1[0] * (1.f / 64.f);
  float var = s2[0] * (1.f / 64.f) - m * m;
  float y = (acc - m) * rsqrtf(var + 1e-3f) * fn_g[oc] + fn_b[oc];
  ysh[oc] = y;
  __syncthreads();
  if (oc < 32) {
    float ov = oc_b[oc];
    for (int dd = 0; dd < 64; ++dd) ov += ysh[dd] * oc_w[dd * 32 + oc];
    outp[(((size_t)b * 48 + i) * 48 + j) * 32 + oc] = ov;
  }
}

#define GRID256(n) (((n) + 255) / 256)

extern "C" void kernel_launch(void* const* d_in, const int* in_sizes, int n_in,
                              void* d_out, int out_size, void* d_ws, size_t ws_size,
                              hipStream_t stream)
{
  const float* x        = (const float*)d_in[0];
  const float* sa_in_w  = (const float*)d_in[1];
  const float* sa_in_b  = (const float*)d_in[2];
  const float* sa_pos_w = (const float*)d_in[3];
  const float* sa_pos_b = (const float*)d_in[4];
  const float* mha_wq   = (const float*)d_in[5];
  const float* mha_bq   = (const float*)d_in[6];
  const float* mha_wk   = (const float*)d_in[7];
  const float* mha_bk   = (const float*)d_in[8];
  const float* mha_wv   = (const float*)d_in[9];
  const float* mha_bv   = (const float*)d_in[10];
  const float* mha_wo   = (const float*)d_in[11];
  const float* mha_bo   = (const float*)d_in[12];
  const float* ln1_g    = (const float*)d_in[13];
  const float* ln1_b    = (const float*)d_in[14];
  const float* ln2_g    = (const float*)d_in[15];
  const float* ln2_b    = (const float*)d_in[16];
  const float* fa_band_w  = (const float*)d_in[17];
  const float* fa_band_b  = (const float*)d_in[18];
  const float* fa_phase_w = (const float*)d_in[19];
  const float* fa_phase_b = (const float*)d_in[20];
  const float* fa_conv_w  = (const float*)d_in[21];
  const float* fa_conv_b  = (const float*)d_in[22];
  const float* wa_app_w   = (const float*)d_in[23];
  const float* wa_app_b   = (const float*)d_in[24];
  const float* wa_det_w3  = (const float*)d_in[25];
  const float* wa_det_b3  = (const float*)d_in[26];
  const float* wa_det_w2  = (const float*)d_in[27];
  const float* wa_det_b2  = (const float*)d_in[28];
  const float* wa_det_w1  = (const float*)d_in[29];
  const float* wa_det_b1  = (const float*)d_in[30];
  const float* fc_w = (const float*)d_in[31];
  const float* fc_b = (const float*)d_in[32];
  const float* fn_g = (const float*)d_in[33];
  const float* fn_b = (const float*)d_in[34];
  const float* oc_w = (const float*)d_in[35];
  const float* oc_b = (const float*)d_in[36];

  float* ws = (float*)d_ws;
  float* out_reg  = (float*)d_out;                 // (2,48,48,32)
  float* spat_reg = out_reg + (size_t)BB * TT * 32;
  float* freq_reg = spat_reg + (size_t)BB * TT * 64;
  float* wav_reg  = freq_reg + (size_t)BB * TT * 64;

  // ---- spatial branch ----
  k_qkv<<<BB * TT, 64, 0, stream>>>(x, sa_in_w, sa_in_b, sa_pos_w, sa_pos_b,
                                    mha_wq, mha_bq, mha_wk, mha_bk, mha_wv, mha_bv,
                                    ws + O_XC, ws + O_Q, ws + O_K, ws + O_V);
  (void)hipFuncSetAttribute((const void*)k_attn,
                            hipFuncAttributeMaxDynamicSharedMemorySize, ATTN_SMEM_BYTES);
  k_attn<<<BHH * QTILES, 128, ATTN_SMEM_BYTES, stream>>>(ws + O_Q, ws + O_K, ws + O_V,
                                                         ws + O_AO);
  k_oproj<<<BB * TT, 64, 0, stream>>>(ws + O_AO, mha_wo, mha_bo, ws + O_XC,
                                      ln1_g, ln1_b, ln2_g, ln2_b, spat_reg);

  // ---- frequency branch ----
  k_dct2_w<<<GRID256(BB * 48 * 48 * 3), 256, 0, stream>>>(x, ws + O_DCT1);
  k_dct2_h<<<GRID256(BB * 48 * 48 * 3), 256, 0, stream>>>(ws + O_DCT1, ws + O_DCT2);
  k_band<<<BB, 256, 0, stream>>>(x, ws + O_DCT2, fa_band_w, fa_band_b,
                                 fa_phase_w, fa_phase_b, ws + O_BW, ws + O_PW);
  k_fconv<<<BB * 48 * 48, 64, 0, stream>>>(ws + O_DCT2, ws + O_BW, ws + O_PW,
                                           fa_conv_w, fa_conv_b, ws + O_FCN);
  k_dct3<<<GRID256(BB * 48 * 48 * 64), 256, 0, stream>>>(ws + O_FCN, ws + O_FDT);
  k_dct3<<<GRID256(BB * 48 * 48 * 64), 256, 0, stream>>>(ws + O_FDT, freq_reg);

  // ---- wavelet branch ----
  k_dwt<<<GRID256((int)SZ1), 256, 0, stream>>>(x, 48, 48,
      ws + O_A1, ws + O_H1, ws + O_V1, ws + O_D1);
  k_dwt<<<GRID256((int)SZ2), 256, 0, stream>>>(ws + O_A1, 24, 24,
      ws + O_A2, ws + O_H2, ws + O_V2, ws + O_D2);
  k_dwt<<<GRID256((int)SZ3), 256, 0, stream>>>(ws + O_A2, 12, 12,
      ws + O_A3, ws + O_H3, ws + O_V3, ws + O_D3);
  k_appw<<<BB, 128, 0, stream>>>(ws + O_A3, wa_app_w, wa_app_b, ws + O_R6);
  k_detw<<<BB * 3, 128, 0, stream>>>(ws + O_H3, ws + O_V3, ws + O_D3,
                                     wa_det_w3, wa_det_b3, 6, ws + O_DW3);
  k_idwt<<<GRID256((int)SZ3), 256, 0, stream>>>(ws + O_R6, ws + O_H3, ws + O_V3, ws + O_D3,
                                                ws + O_DW3, 6, ws + O_R12);
  k_detw<<<BB * 3, 128, 0, stream>>>(ws + O_H2, ws + O_V2, ws + O_D2,
                                     wa_det_w2, wa_det_b2, 12, ws + O_DW2);
  k_idwt<<<GRID256((int)SZ2), 256, 0, stream>>>(ws + O_R12, ws + O_H2, ws + O_V2, ws + O_D2,
                                                ws + O_DW2, 12, ws + O_R24);
  k_detw<<<BB * 3, 128, 0, stream>>>(ws + O_H1, ws + O_V1, ws + O_D1,
                                     wa_det_w1, wa_det_b1, 24, ws + O_DW1);
  k_idwt<<<GRID256((int)SZ1), 256, 0, stream>>>(ws + O_R24, ws + O_H1, ws + O_V1, ws + O_D1,
                                                ws + O_DW1, 24, wav_reg);

  // ---- combine ----
  k_combine<<<BB * 48 * 48, 64, 0, stream>>>(spat_reg, freq_reg, wav_reg,
                                             fc_w, fc_b, fn_g, fn_b, oc_w, oc_b, out_reg);
}